// MyRNNCell_80221399154914
// MI455X (gfx1250) — compile-verified
//
#include <hip/hip_runtime.h>

// ---------------------------------------------------------------------------
// Problem constants (match reference)
// ---------------------------------------------------------------------------
constexpr int NN      = 4096;          // vector length / matrix dim
constexpr int NPAIR   = NN / 8;        // 512 K-pairs (2 WMMA k=4 chunks each)
constexpr int KSEG    = 32;            // K split for parallelism -> 8192 waves
constexpr int PPSEG   = NPAIR / KSEG;  // 16 pairs per wave
constexpr int NRB     = NN / 16;       // 256 row blocks of 16
constexpr int NSTEPS  = 256;
constexpr int VSTRIDE = NPAIR * 128;   // floats per B-pack variant (65536)

typedef __attribute__((ext_vector_type(2))) float v2f;
typedef __attribute__((ext_vector_type(4))) float v4f;
typedef __attribute__((ext_vector_type(8))) float v8f;

__device__ __forceinline__ v2f lo2(v4f x) { return __builtin_shufflevector(x, x, 0, 1); }
__device__ __forceinline__ v2f hi2(v4f x) { return __builtin_shufflevector(x, x, 2, 3); }

// ws layout in floats
constexpr size_t WS_BPACK = 0;                        // 4*65536   = 262144
constexpr size_t WS_PART  = WS_BPACK + 4u * VSTRIDE;  // 256*32*32 = 262144
constexpr size_t WS_ARE   = WS_PART + (size_t)NRB * KSEG * 32;
constexpr size_t WS_AIM   = WS_ARE + NN;

// ---------------------------------------------------------------------------
// 1) Pack B operands, permuted-K layout matched to b128 A loads.
//    Per pair p, lane L holds a contiguous float4: logical k = 8p + 4*(L>>4) + j.
//    WMMA#1 of the pair consumes elements {0,1}, WMMA#2 elements {2,3}.
//    flat idx = v*VSTRIDE + p*128 + L*4 + j   (per-lane quads, coalesced b128)
//    variant 0: A=U_re -> col0=z_re,  col1=z_im
//    variant 1: A=U_im -> col0=-z_im, col1=z_re
//    variant 2: A=V_re -> col0=x_re,  col1=x_im
//    variant 3: A=V_im -> col0=-x_im, col1=x_re
// ---------------------------------------------------------------------------
__global__ void pack_b_kernel(const float* __restrict__ z_re,
                              const float* __restrict__ z_im,
                              const float* __restrict__ x_re,
                              const float* __restrict__ x_im,
                              float* __restrict__ bpack) {
  int g = blockIdx.x * blockDim.x + threadIdx.x;   // < 262144
  int j = g & 3;
  int L = (g >> 2) & 31;
  int p = (g >> 7) & (NPAIR - 1);
  int v = g >> 16;                                  // 0..3
  int n = L & 15;
  int k = 8 * p + 4 * (L >> 4) + j;

  float val = 0.0f;
  if (v == 0)      val = (n == 0) ?  z_re[k] : (n == 1) ? z_im[k] : 0.0f;
  else if (v == 1) val = (n == 0) ? -z_im[k] : (n == 1) ? z_re[k] : 0.0f;
  else if (v == 2) val = (n == 0) ?  x_re[k] : (n == 1) ? x_im[k] : 0.0f;
  else             val = (n == 0) ? -x_im[k] : (n == 1) ? x_re[k] : 0.0f;
  bpack[g] = val;
}

// ---------------------------------------------------------------------------
// 2) HBM-bound GEMV via V_WMMA_F32_16X16X4_F32, b128 streaming loads.
//    One wave = (row block rb, K segment ks). Per pair: 8 non-temporal b128
//    matrix loads? no -- 4 matrix loads + 4 B loads, then 8 chained WMMAs.
// ---------------------------------------------------------------------------
__global__ void gemv_wmma_kernel(const float* __restrict__ Ure,
                                 const float* __restrict__ Uim,
                                 const float* __restrict__ Vre,
                                 const float* __restrict__ Vim,
                                 const float* __restrict__ bpack,
                                 float* __restrict__ part) {
  const int wave  = threadIdx.x >> 5;
  const int lane  = threadIdx.x & 31;
  const int task  = blockIdx.x * 8 + wave;     // 0..8191
  const int rb    = task >> 5;                 // 0..255
  const int ks    = task & (KSEG - 1);         // 0..31
  const int row   = rb * 16 + (lane & 15);
  const int khalf = lane >> 4;
  const int p0    = ks * PPSEG;

  const float* pUre = Ure + (size_t)row * NN + 8 * p0 + 4 * khalf;
  const float* pUim = Uim + (size_t)row * NN + 8 * p0 + 4 * khalf;
  const float* pVre = Vre + (size_t)row * NN + 8 * p0 + 4 * khalf;
  const float* pVim = Vim + (size_t)row * NN + 8 * p0 + 4 * khalf;
  const float* pB   = bpack + ((size_t)p0 * 32 + lane) * 4;

  v8f acc = {};

#pragma unroll 2
  for (int p = 0; p < PPSEG; ++p) {
    v4f aur = __builtin_nontemporal_load((const v4f*)pUre);  pUre += 8;
    v4f aui = __builtin_nontemporal_load((const v4f*)pUim);  pUim += 8;
    v4f avr = __builtin_nontemporal_load((const v4f*)pVre);  pVre += 8;
    v4f avi = __builtin_nontemporal_load((const v4f*)pVim);  pVim += 8;
    v4f bz  = *(const v4f*)(pB);
    v4f bzi = *(const v4f*)(pB + VSTRIDE);
    v4f bx  = *(const v4f*)(pB + 2 * VSTRIDE);
    v4f bxi = *(const v4f*)(pB + 3 * VSTRIDE);
    pB += 128;

    acc = __builtin_amdgcn_wmma_f32_16x16x4_f32(false, lo2(aur), false, lo2(bz),
                                                (short)0, acc, false, false);
    acc = __builtin_amdgcn_wmma_f32_16x16x4_f32(false, hi2(aur), false, hi2(bz),
                                                (short)0, acc, false, false);
    acc = __builtin_amdgcn_wmma_f32_16x16x4_f32(false, lo2(aui), false, lo2(bzi),
                                                (short)0, acc, false, false);
    acc = __builtin_amdgcn_wmma_f32_16x16x4_f32(false, hi2(aui), false, hi2(bzi),
                                                (short)0, acc, false, false);
    acc = __builtin_amdgcn_wmma_f32_16x16x4_f32(false, lo2(avr), false, lo2(bx),
                                                (short)0, acc, false, false);
    acc = __builtin_amdgcn_wmma_f32_16x16x4_f32(false, hi2(avr), false, hi2(bx),
                                                (short)0, acc, false, false);
    acc = __builtin_amdgcn_wmma_f32_16x16x4_f32(false, lo2(avi), false, lo2(bxi),
                                                (short)0, acc, false, false);
    acc = __builtin_amdgcn_wmma_f32_16x16x4_f32(false, hi2(avi), false, hi2(bxi),
                                                (short)0, acc, false, false);
  }

  // C layout: VGPR r, lanes0-15 -> (M=r, N=lane); lanes16-31 -> (M=r+8, N=lane-16)
  // Useful columns: n==0 (Re), n==1 (Im). Partial rec: [re m0..15 | im m0..15]
  int n = lane & 15;
  if (n < 2) {
    union { v8f v; float f[8]; } u; u.v = acc;
    float* dst = part + ((size_t)rb * KSEG + ks) * 32 + n * 16 + khalf * 8;
#pragma unroll
    for (int r = 0; r < 8; ++r) dst[r] = u.f[r];
  }
}

// ---------------------------------------------------------------------------
// 3) Deterministic reduction of K-segment partials -> a_re, a_im
// ---------------------------------------------------------------------------
__global__ void reduce_a_kernel(const float* __restrict__ part,
                                float* __restrict__ a_re,
                                float* __restrict__ a_im) {
  int t = blockIdx.x * blockDim.x + threadIdx.x;   // < 8192
  int comp = t >> 12;                              // 0=re, 1=im
  int row  = t & (NN - 1);
  int rb = row >> 4, m = row & 15;
  float s = 0.0f;
#pragma unroll
  for (int ks = 0; ks < KSEG; ++ks)
    s += part[((size_t)rb * KSEG + ks) * 32 + comp * 16 + m];
  if (comp) a_im[row] = s; else a_re[row] = s;
}

// ---------------------------------------------------------------------------
// 4) Elementwise DoPri5 fixed-step integration of dz/dt = (a - |z|^2) z
// ---------------------------------------------------------------------------
__global__ void ode_kernel(const float* __restrict__ a_re,
                           const float* __restrict__ a_im,
                           const float* __restrict__ z_re0,
                           const float* __restrict__ z_im0,
                           float* __restrict__ out) {
  int i = blockIdx.x * blockDim.x + threadIdx.x;   // < 4096
  const float ar = a_re[i], ai = a_im[i];
  float zr = z_re0[i], zi = z_im0[i];
  const float h = 6.28318530717958647692f / (float)NSTEPS;

  auto f = [&](float r, float m, float& fr, float& fi) {
    float mag = r * r + m * m;
    float wr = ar - mag;
    fr = wr * r - ai * m;
    fi = wr * m + ai * r;
  };

  for (int s = 0; s < NSTEPS; ++s) {
    float k1r, k1i, k2r, k2i, k3r, k3i, k4r, k4i, k5r, k5i, k6r, k6i;
    f(zr, zi, k1r, k1i);
    f(zr + h * (0.2f * k1r), zi + h * (0.2f * k1i), k2r, k2i);
    f(zr + h * (0.075f * k1r + 0.225f * k2r),
      zi + h * (0.075f * k1i + 0.225f * k2i), k3r, k3i);
    f(zr + h * ((44.0f/45.0f) * k1r - (56.0f/15.0f) * k2r + (32.0f/9.0f) * k3r),
      zi + h * ((44.0f/45.0f) * k1i - (56.0f/15.0f) * k2i + (32.0f/9.0f) * k3i),
      k4r, k4i);
    f(zr + h * ((19372.0f/6561.0f) * k1r - (25360.0f/2187.0f) * k2r
              + (64448.0f/6561.0f) * k3r - (212.0f/729.0f) * k4r),
      zi + h * ((19372.0f/6561.0f) * k1i - (25360.0f/2187.0f) * k2i
              + (64448.0f/6561.0f) * k3i - (212.0f/729.0f) * k4i),
      k5r, k5i);
    f(zr + h * ((9017.0f/3168.0f) * k1r - (355.0f/33.0f) * k2r
              + (46732.0f/5247.0f) * k3r + (49.0f/176.0f) * k4r
              - (5103.0f/18656.0f) * k5r),
      zi + h * ((9017.0f/3168.0f) * k1i - (355.0f/33.0f) * k2i
              + (46732.0f/5247.0f) * k3i + (49.0f/176.0f) * k4i
              - (5103.0f/18656.0f) * k5i),
      k6r, k6i);
    zr = zr + h * ((35.0f/384.0f) * k1r + (500.0f/1113.0f) * k3r
                 + (125.0f/192.0f) * k4r - (2187.0f/6784.0f) * k5r
                 + (11.0f/84.0f) * k6r);
    zi = zi + h * ((35.0f/384.0f) * k1i + (500.0f/1113.0f) * k3i
                 + (125.0f/192.0f) * k4i - (2187.0f/6784.0f) * k5i
                 + (11.0f/84.0f) * k6i);
  }
  out[i]      = zr;   // real part  (2, N, 1) stacked
  out[NN + i] = zi;   // imag part
}

// ---------------------------------------------------------------------------
// Launch: inputs order = x_re, x_im, z_re, z_im, U_re, U_im, V_re, V_im
// ---------------------------------------------------------------------------
extern "C" void kernel_launch(void* const* d_in, const int* in_sizes, int n_in,
                              void* d_out, int out_size, void* d_ws, size_t ws_size,
                              hipStream_t stream) {
  const float* x_re = (const float*)d_in[0];
  const float* x_im = (const float*)d_in[1];
  const float* z_re = (const float*)d_in[2];
  const float* z_im = (const float*)d_in[3];
  const float* U_re = (const float*)d_in[4];
  const float* U_im = (const float*)d_in[5];
  const float* V_re = (const float*)d_in[6];
  const float* V_im = (const float*)d_in[7];
  float* out = (float*)d_out;

  float* ws    = (float*)d_ws;
  float* bpack = ws + WS_BPACK;
  float* part  = ws + WS_PART;
  float* a_re  = ws + WS_ARE;
  float* a_im  = ws + WS_AIM;

  pack_b_kernel<<<1024, 256, 0, stream>>>(z_re, z_im, x_re, x_im, bpack);
  gemv_wmma_kernel<<<1024, 256, 0, stream>>>(U_re, U_im, V_re, V_im, bpack, part);
  reduce_a_kernel<<<32, 256, 0, stream>>>(part, a_re, a_im);
  ode_kernel<<<16, 256, 0, stream>>>(a_re, a_im, z_re, z_im, out);
}